// NeighLayer_36644660969839
// MI455X (gfx1250) — compile-verified
//
#include <hip/hip_runtime.h>

#define N_NODES 40000
#define N_EDGES 640000
#define D_FEAT  128
#define EPB     256   // edges per block (N_EDGES % EPB == 0)

// ---------------- gather * a -> scatter-add (defined first so the disasm
// snippet shows this kernel's body) ----------------
__global__ __launch_bounds__(256) void gnn_scatter_kernel(
    const float* __restrict__ x,
    const int*   __restrict__ rows,
    const int*   __restrict__ cols,
    const float* __restrict__ vals,
    float*       __restrict__ agg,   // == d_out, pre-zeroed
    float*       __restrict__ deg) {

    __shared__ int   sRow[EPB];
    __shared__ int   sCol[EPB];
    __shared__ float sVal[EPB];

    const int t    = threadIdx.x;
    const int base = blockIdx.x * EPB;

    // Stage this block's edge metadata into LDS via the async-copy engine
    // (GLOBAL_LOAD_ASYNC_TO_LDS_B32, tracked by ASYNCcnt). vdst = LDS byte
    // offset (low 32 bits of the generic pointer to a __shared__ object),
    // vaddr = 64-bit global address.
    {
        unsigned lRow = (unsigned)(uintptr_t)&sRow[t];
        unsigned lCol = (unsigned)(uintptr_t)&sCol[t];
        unsigned lVal = (unsigned)(uintptr_t)&sVal[t];
        const int*   gRow = rows + base + t;
        const int*   gCol = cols + base + t;
        const float* gVal = vals + base + t;
        asm volatile(
            "global_load_async_to_lds_b32 %0, %3, off\n\t"
            "global_load_async_to_lds_b32 %1, %4, off\n\t"
            "global_load_async_to_lds_b32 %2, %5, off\n\t"
            "s_wait_asynccnt 0"
            :
            : "v"(lRow), "v"(lCol), "v"(lVal),
              "v"(gRow), "v"(gCol), "v"(gVal)
            : "memory");
    }
    __syncthreads();

    const int wave = t >> 5;   // 8 waves per block, one edge per wave per round
    const int lane = t & 31;   // wave32: lane handles 4 contiguous floats

    #pragma unroll 4
    for (int r = 0; r < EPB / 8; ++r) {
        const int   e   = r * 8 + wave;
        const int   row = sRow[e];          // broadcast LDS reads (no bank conflict)
        const int   col = sCol[e];
        const float a   = sVal[e];

        // Prefetch the gather row two rounds ahead into L0 (global_prefetch_b8).
        if (r + 2 < EPB / 8) {
            const int c2 = sCol[(r + 2) * 8 + wave];
            __builtin_prefetch(x + (size_t)c2 * D_FEAT + lane * 4, 0, 1);
        }

        const float4 v = ((const float4*)(x + (size_t)col * D_FEAT))[lane];
        float* dst = agg + (size_t)row * D_FEAT + lane * 4;

        // Hardware f32 atomics into L2-resident accumulator.
        unsafeAtomicAdd(dst + 0, a * v.x);
        unsafeAtomicAdd(dst + 1, a * v.y);
        unsafeAtomicAdd(dst + 2, a * v.z);
        unsafeAtomicAdd(dst + 3, a * v.w);

        if (lane == 0) unsafeAtomicAdd(deg + row, a);
    }
}

// ---------------- zero accumulators ----------------
__global__ __launch_bounds__(256) void gnn_zero_kernel(float4* __restrict__ out4,
                                                       float* __restrict__ deg,
                                                       int n4, int ndeg) {
    int i = blockIdx.x * blockDim.x + threadIdx.x;
    if (i < n4)   out4[i] = make_float4(0.f, 0.f, 0.f, 0.f);
    if (i < ndeg) deg[i] = 0.f;
}

// ---------------- out = (deg==0 ? 0 : out/deg) + bias ----------------
__global__ __launch_bounds__(256) void gnn_finalize_kernel(
    float4* __restrict__ out4,
    const float* __restrict__ deg,
    const float4* __restrict__ bias4) {

    const int i = blockIdx.x * blockDim.x + threadIdx.x; // float4 index
    const int n  = i >> 5;        // 32 float4 per 128-float row
    const int f4 = i & 31;
    if (n >= N_NODES) return;

    const float  d = deg[n];      // broadcast within wave
    const float4 b = bias4[f4];
    float4 v = out4[i];

    if (d == 0.f) {
        v = b;
    } else {
        v.x = v.x / d + b.x;
        v.y = v.y / d + b.y;
        v.z = v.z / d + b.z;
        v.w = v.w / d + b.w;
    }
    out4[i] = v;
}

extern "C" void kernel_launch(void* const* d_in, const int* in_sizes, int n_in,
                              void* d_out, int out_size, void* d_ws, size_t ws_size,
                              hipStream_t stream) {
    const float* x    = (const float*)d_in[0];
    const int*   rows = (const int*)  d_in[1];
    const int*   cols = (const int*)  d_in[2];
    const float* vals = (const float*)d_in[3];
    const float* bias = (const float*)d_in[4];
    float* out = (float*)d_out;
    float* deg = (float*)d_ws;               // 40000 floats of scratch

    const int n4 = N_NODES * D_FEAT / 4;     // 1,280,000 float4

    gnn_zero_kernel<<<(n4 + 255) / 256, 256, 0, stream>>>(
        (float4*)out, deg, n4, N_NODES);

    gnn_scatter_kernel<<<N_EDGES / EPB, 256, 0, stream>>>(
        x, rows, cols, vals, out, deg);

    gnn_finalize_kernel<<<(n4 + 255) / 256, 256, 0, stream>>>(
        (float4*)out, deg, (const float4*)bias);
}